// QuantizedNeuralNetwork_4312147165524
// MI455X (gfx1250) — compile-verified
//
#include <hip/hip_runtime.h>

typedef float v2f __attribute__((ext_vector_type(2)));
typedef float v8f __attribute__((ext_vector_type(8)));

#define IN_F   1024
#define OUT_F  256
#define BATCH  256

// ---------------------------------------------------------------------------
// Kernel 1: Gram matrices via f32 WMMA.
//   G[i,j] = sum_b qX[b,i] * wX[b,j]
//   H[i,j] = sum_b qX[b,i] * qX[b,j]
// One wave32 per 16x16 output tile, shared A operand (qX^T tile), two WMMA
// accumulators. K (=batch=256) consumed 4 at a time by V_WMMA_F32_16X16X4_F32.
//
// 32-bit A 16x4 layout (ISA 7.12.2): lanes 0-15 -> M=lane, VGPR0=K0,VGPR1=K1;
// lanes 16-31 -> M=lane-16, VGPR0=K2, VGPR1=K3. B 4x16 mirrors the K split.
// ---------------------------------------------------------------------------
__global__ __launch_bounds__(256)
void gram_wmma_kernel(const float* __restrict__ qX,
                      const float* __restrict__ wX,
                      float* __restrict__ G,
                      float* __restrict__ H)
{
    const int wave = threadIdx.x >> 5;
    const int lane = threadIdx.x & 31;
    const int tile = blockIdx.x * (blockDim.x >> 5) + wave;   // 0..4095
    const int i0 = (tile >> 6) << 4;   // output row block  (0..1008)
    const int j0 = (tile & 63) << 4;   // output col block
    const int l16   = lane & 15;
    const int khalf = (lane >> 4) << 1;  // 0 for lanes 0-15, 2 for lanes 16-31

    v8f cg = {};   // G accumulator
    v8f ch = {};   // H accumulator

    for (int b0 = 0; b0 < BATCH; b0 += 4) {
        const int k0 = b0 + khalf;          // this lane's first K index
        v2f a, bw, bq;
        // A tile (qX^T rows i0..i0+15): A[M=l16][K] = qX[K*IN_F + i0 + l16]
        a.x  = qX[(k0    ) * IN_F + i0 + l16];
        a.y  = qX[(k0 + 1) * IN_F + i0 + l16];
        // B tiles: B[K][N=l16] = X[K*IN_F + j0 + l16]
        bw.x = wX[(k0    ) * IN_F + j0 + l16];
        bw.y = wX[(k0 + 1) * IN_F + j0 + l16];
        bq.x = qX[(k0    ) * IN_F + j0 + l16];
        bq.y = qX[(k0 + 1) * IN_F + j0 + l16];

        // 8 args: (neg_a, A, neg_b, B, c_mod, C, reuse_a, reuse_b)
        cg = __builtin_amdgcn_wmma_f32_16x16x4_f32(false, a, false, bw,
                                                   (short)0, cg, false, false);
        ch = __builtin_amdgcn_wmma_f32_16x16x4_f32(false, a, false, bq,
                                                   (short)0, ch, false, false);
    }

    // C/D layout: VGPR v -> row v (lanes 0-15) or row v+8 (lanes 16-31), col = l16
    const int mbase = (lane < 16) ? 0 : 8;
#pragma unroll
    for (int v = 0; v < 8; ++v) {
        const int row = i0 + mbase + v;
        const int col = j0 + l16;
        G[row * IN_F + col] = cg[v];
        H[row * IN_F + col] = ch[v];
    }
}

// ---------------------------------------------------------------------------
// Kernel 2: the (now short) sequential core. Single workgroup, persistent.
//   acc[i] (LDS) tracks  c_i = qX[:,i] . u  for the current scan position.
//   Per row: 32 blocks of 32 columns.
//     - wave 0 runs the in-block triangular scan (lane i owns column j0+i,
//       acc_j broadcast by shuffle, in-register rank-1 updates)
//     - all 8 waves then apply the rank-32 panel update to acc[0..1023]
//       with b128 streaming loads of the L2-resident G/H panels; after the
//       full row, acc equals the next row's carry  c <- c + G w - H q.
// ---------------------------------------------------------------------------
__global__ __launch_bounds__(256)
void gptq_seq_kernel(const float* __restrict__ W,
                     const float* __restrict__ G,
                     const float* __restrict__ H,
                     const float* __restrict__ alphabet,
                     float* __restrict__ Q)
{
    __shared__ float acc[IN_F];
    __shared__ float wrow[IN_F];
    __shared__ float qrow[IN_F];
    __shared__ float Gblk[32][32];
    __shared__ float Hblk[32][32];

    const int tid  = threadIdx.x;
    const int lane = tid & 31;
    const int wave = tid >> 5;

    const float a0 = alphabet[0];
    const float a1 = alphabet[1];
    const float a2 = alphabet[2];

    for (int i = tid; i < IN_F; i += 256) acc[i] = 0.0f;   // u0 = 0  ->  c = 0
    __syncthreads();

    for (int r = 0; r < OUT_F; ++r) {
        // W row: 1024 floats = 256 threads x one b128 load
        ((float4*)wrow)[tid] = ((const float4*)(W + r * IN_F))[tid];
        __syncthreads();

        for (int b = 0; b < 32; ++b) {
            const int j0 = b << 5;

            // stage the 32x32 diagonal blocks of G and H: one float4 per thread
            {
                const int ii  = tid >> 3;        // 0..31
                const int jj4 = (tid & 7) << 2;  // 0,4,...,28
                *(float4*)&Gblk[ii][jj4] =
                    *(const float4*)&G[(j0 + ii) * IN_F + (j0 + jj4)];
                *(float4*)&Hblk[ii][jj4] =
                    *(const float4*)&H[(j0 + ii) * IN_F + (j0 + jj4)];
            }
            __syncthreads();

            if (wave == 0) {
                float av = acc[j0 + lane];   // includes all j < j0 contributions
                float qj_store = 0.0f;
#pragma unroll 1
                for (int j = 0; j < 32; ++j) {
                    const float dot_u = __shfl(av, j, 32);
                    const float wj    = wrow[j0 + j];
                    const float nsq   = Hblk[j][j];        // qX[:,j].qX[:,j]
                    const float cross = Gblk[j][j];        // qX[:,j].wX[:,j]
                    const bool  degen = nsq < 1e-32f;      // sqrt(nsq) < 1e-16
                    const float safe  = degen ? 1.0f : nsq;
                    const float proj  = (dot_u + wj * cross) / safe;
                    const float target = (fabsf(dot_u) < 1e-10f) ? wj : proj;
                    // argmin over the 3-letter alphabet (first-min tie-break)
                    float best = fabsf(a0 - target);
                    float q    = a0;
                    const float d1 = fabsf(a1 - target);
                    if (d1 < best) { best = d1; q = a1; }
                    const float d2 = fabsf(a2 - target);
                    if (d2 < best) { best = d2; q = a2; }
                    q = degen ? 0.0f : q;
                    if (lane == j) qj_store = q;
                    if (lane > j)  av += wj * Gblk[lane][j] - q * Hblk[lane][j];
                }
                qrow[j0 + lane]         = qj_store;
                Q[r * IN_F + j0 + lane] = qj_store;
            }
            __syncthreads();

            // rank-32 panel update over ALL columns:
            //   acc[i] += sum_{j in block} ( w_j G[i,j] - q_j H[i,j] )
            // b128 streaming loads: 8 G-quads + 8 H-quads per (thread, i).
            for (int i = tid; i < IN_F; i += 256) {
                const float4* gp4 = (const float4*)&G[i * IN_F + j0];
                const float4* hp4 = (const float4*)&H[i * IN_F + j0];
                if (i + 256 < IN_F) {
                    __builtin_prefetch(&G[(i + 256) * IN_F + j0], 0, 1);
                    __builtin_prefetch(&H[(i + 256) * IN_F + j0], 0, 1);
                }
                float s = 0.0f;
#pragma unroll
                for (int j4 = 0; j4 < 8; ++j4) {
                    const float4 g = gp4[j4];
                    const float4 h = hp4[j4];
                    const int jb = j0 + (j4 << 2);
                    s += wrow[jb + 0] * g.x - qrow[jb + 0] * h.x;
                    s += wrow[jb + 1] * g.y - qrow[jb + 1] * h.y;
                    s += wrow[jb + 2] * g.z - qrow[jb + 2] * h.z;
                    s += wrow[jb + 3] * g.w - qrow[jb + 3] * h.w;
                }
                acc[i] += s;
            }
            __syncthreads();
        }
    }
}

// ---------------------------------------------------------------------------
// Launch: inputs in setup_inputs() order: W, wX, qX, alphabet (all f32).
// d_out: 256*1024 f32. Workspace: G (4MB) then H (4MB).
// ---------------------------------------------------------------------------
extern "C" void kernel_launch(void* const* d_in, const int* in_sizes, int n_in,
                              void* d_out, int out_size, void* d_ws, size_t ws_size,
                              hipStream_t stream)
{
    const float* W        = (const float*)d_in[0];
    const float* wX       = (const float*)d_in[1];
    const float* qX       = (const float*)d_in[2];
    const float* alphabet = (const float*)d_in[3];
    float* Q = (float*)d_out;

    float* G = (float*)d_ws;
    float* H = G + (size_t)IN_F * IN_F;

    // 64*64 = 4096 16x16 tiles, 8 waves per 256-thread block -> 512 blocks
    gram_wmma_kernel<<<512, 256, 0, stream>>>(qX, wX, G, H);

    // single persistent workgroup runs the (shortened) sequential recurrence
    gptq_seq_kernel<<<1, 256, 0, stream>>>(W, G, H, alphabet, Q);
}